// FastfoodLayer_4904852652182
// MI455X (gfx1250) — compile-verified
//
#include <hip/hip_runtime.h>
#include <math.h>

// Fastfood layer for MI455X (gfx1250, wave32).
// FWHT_1024 = (H4 (x) H16) * X * H16 with X the row reshaped 64x16.
// Stages 1 and 2 run on the matrix pipe as chained V_WMMA_F32_16X16X4_F32.
// v2: __cosf epilogue (v_cos_f32), B128-vectorized LDS/global traffic,
//     LDS row stride 20 (conflict-free for column access, 16B-aligned rows).

typedef __attribute__((ext_vector_type(2))) float v2f;
typedef __attribute__((ext_vector_type(8))) float v8f;

#define DD    1024
#define MMB   4
#define NROWS 8192
#define LSTR  20      // padded LDS row stride: 20l mod 64 distinct for l=0..15,
                      // and 20*4B = 80B row base keeps 16B alignment for b128 ops

__device__ __forceinline__ float hsign(int j, int k) {
    // Sylvester Hadamard entry: H[j,k] = (-1)^popc(j&k)
    return (__popc(j & k) & 1) ? -1.0f : 1.0f;
}

// FWHT over a 64x16 (row-stride LSTR) LDS matrix; buf is input + scratch,
// result -> out. 4 waves per block; wave t owns tile rows [16t,16t+16).
__device__ __forceinline__ void fwht1024_wmma(float* __restrict__ buf,
                                              float* __restrict__ out,
                                              int wave, int lane, int tid) {
    const int half = lane >> 4;    // 0: lanes 0-15, 1: lanes 16-31
    const int l    = lane & 15;
    const int t    = wave;

    // ---- stage 1: Z_t = X_t @ H16  (4 chained 16x16x4 f32 WMMAs) ----
    v8f acc = {0.f,0.f,0.f,0.f,0.f,0.f,0.f,0.f};
#pragma unroll
    for (int k = 0; k < 4; ++k) {
        const int c0 = 4 * k + 2 * half;            // A: K={0,1}|{2,3} per half
        v2f a, b;
        a.x = buf[(16 * t + l) * LSTR + c0];
        a.y = buf[(16 * t + l) * LSTR + c0 + 1];
        b.x = hsign(c0,     l);                     // B: H16 rows striped over lanes
        b.y = hsign(c0 + 1, l);
        acc = __builtin_amdgcn_wmma_f32_16x16x4_f32(false, a, false, b,
                                                    (short)0, acc, false, false);
    }
#pragma unroll
    for (int r = 0; r < 8; ++r)                     // C/D: VGPR r = rows r, r+8
        buf[(16 * t + r + 8 * half) * LSTR + l] = acc[r];
    __syncthreads();

    // ---- stage 2: W_t = H16 @ Z_t ----
    v8f acc2 = {0.f,0.f,0.f,0.f,0.f,0.f,0.f,0.f};
#pragma unroll
    for (int k = 0; k < 4; ++k) {
        const int r0 = 4 * k + 2 * half;
        v2f a, b;
        a.x = hsign(l, r0);                         // A = H16 columns chunk
        a.y = hsign(l, r0 + 1);
        b.x = buf[(16 * t + r0) * LSTR + l];        // B = Z_t rows chunk
        b.y = buf[(16 * t + r0 + 1) * LSTR + l];
        acc2 = __builtin_amdgcn_wmma_f32_16x16x4_f32(false, a, false, b,
                                                     (short)0, acc2, false, false);
    }
#pragma unroll
    for (int r = 0; r < 8; ++r)
        buf[(16 * t + r + 8 * half) * LSTR + l] = acc2[r];
    __syncthreads();

    // ---- stage 3: H4 combine across the 4 tiles, b128 LDS traffic ----
    {
        const int jrow = tid >> 1;                  // 0..63
        const int hc   = (tid & 1) * 8;             // column half: 0 or 8
        const int a4   = jrow >> 4;
        const int b4   = jrow & 15;
        float4 s0 = {0.f,0.f,0.f,0.f}, s1 = {0.f,0.f,0.f,0.f};
#pragma unroll
        for (int ap = 0; ap < 4; ++ap) {
            const float  sg = hsign(a4, ap);
            const float4 w0 = *(const float4*)&buf[(ap * 16 + b4) * LSTR + hc];
            const float4 w1 = *(const float4*)&buf[(ap * 16 + b4) * LSTR + hc + 4];
            s0.x += sg * w0.x; s0.y += sg * w0.y; s0.z += sg * w0.z; s0.w += sg * w0.w;
            s1.x += sg * w1.x; s1.y += sg * w1.y; s1.z += sg * w1.z; s1.w += sg * w1.w;
        }
        *(float4*)&out[jrow * LSTR + hc]     = s0;
        *(float4*)&out[jrow * LSTR + hc + 4] = s1;
    }
    __syncthreads();
}

__global__ __launch_bounds__(128)
void fastfood_wmma_kernel(const float* __restrict__ x, const float* __restrict__ B,
                          const float* __restrict__ G, const float* __restrict__ S,
                          const int* __restrict__ P, float* __restrict__ out) {
    __shared__ __align__(16) float xrow[DD];
    __shared__ __align__(16) float bufA[64 * LSTR];
    __shared__ __align__(16) float bufB[64 * LSTR];

    const int row  = blockIdx.x;
    const int tid  = threadIdx.x;
    const int wave = tid >> 5;
    const int lane = tid & 31;

    // this thread's 8 consecutive elements: j0..j0+7 (one LDS row half)
    const int j0   = tid * 8;
    const int jrow = tid >> 1;            // j0 >> 4
    const int hc   = (tid & 1) * 8;       // j0 & 15
    const int lidx = jrow * LSTR + hc;    // 16B-aligned LDS offset

    // stage row of x into LDS once; reused by all 4 m-blocks (4x less HBM read)
    *(float4*)&xrow[j0]     = *(const float4*)&x[(size_t)row * DD + j0];
    *(float4*)&xrow[j0 + 4] = *(const float4*)&x[(size_t)row * DD + j0 + 4];
    __syncthreads();

    for (int m = 0; m < MMB; ++m) {
        const float* Bm = B + m * DD;
        const float* Gm = G + m * DD;
        const float* Sm = S + m * DD;
        const int*   Pm = P + m * DD;

        // w1 = x * B[m]
        {
            const float4 xv0 = *(const float4*)&xrow[j0];
            const float4 xv1 = *(const float4*)&xrow[j0 + 4];
            const float4 bv0 = *(const float4*)&Bm[j0];
            const float4 bv1 = *(const float4*)&Bm[j0 + 4];
            float4 w0, w1;
            w0.x = xv0.x * bv0.x; w0.y = xv0.y * bv0.y;
            w0.z = xv0.z * bv0.z; w0.w = xv0.w * bv0.w;
            w1.x = xv1.x * bv1.x; w1.y = xv1.y * bv1.y;
            w1.z = xv1.z * bv1.z; w1.w = xv1.w * bv1.w;
            *(float4*)&bufA[lidx]     = w0;
            *(float4*)&bufA[lidx + 4] = w1;
        }
        __syncthreads();

        fwht1024_wmma(bufA, bufB, wave, lane, tid);       // h1 -> bufB

        // permute + multiply by G[m]  (gather is inherently scalar b32 reads)
        {
            const int4   p0 = *(const int4*)&Pm[j0];
            const int4   p1 = *(const int4*)&Pm[j0 + 4];
            const float4 g0 = *(const float4*)&Gm[j0];
            const float4 g1 = *(const float4*)&Gm[j0 + 4];
            float4 w0, w1;
            w0.x = bufB[(p0.x >> 4) * LSTR + (p0.x & 15)] * g0.x;
            w0.y = bufB[(p0.y >> 4) * LSTR + (p0.y & 15)] * g0.y;
            w0.z = bufB[(p0.z >> 4) * LSTR + (p0.z & 15)] * g0.z;
            w0.w = bufB[(p0.w >> 4) * LSTR + (p0.w & 15)] * g0.w;
            w1.x = bufB[(p1.x >> 4) * LSTR + (p1.x & 15)] * g1.x;
            w1.y = bufB[(p1.y >> 4) * LSTR + (p1.y & 15)] * g1.y;
            w1.z = bufB[(p1.z >> 4) * LSTR + (p1.z & 15)] * g1.z;
            w1.w = bufB[(p1.w >> 4) * LSTR + (p1.w & 15)] * g1.w;
            *(float4*)&bufA[lidx]     = w0;
            *(float4*)&bufA[lidx + 4] = w1;
        }
        __syncthreads();

        fwht1024_wmma(bufA, bufB, wave, lane, tid);       // h2 -> bufB

        // v = S*h2/sqrt(1024); out = cos(v) -> v_cos_f32 (fast path; |v| is
        // O(100) rad, f32 revolution-domain error ~1e-4 rad, well in tolerance)
        {
            const float inv = 0.03125f;                    // 1/32
            const float4 s0 = *(const float4*)&Sm[j0];
            const float4 s1 = *(const float4*)&Sm[j0 + 4];
            const float4 h0 = *(const float4*)&bufB[lidx];
            const float4 h1 = *(const float4*)&bufB[lidx + 4];
            float4 o0, o1;
            o0.x = __cosf(s0.x * h0.x * inv);
            o0.y = __cosf(s0.y * h0.y * inv);
            o0.z = __cosf(s0.z * h0.z * inv);
            o0.w = __cosf(s0.w * h0.w * inv);
            o1.x = __cosf(s1.x * h1.x * inv);
            o1.y = __cosf(s1.y * h1.y * inv);
            o1.z = __cosf(s1.z * h1.z * inv);
            o1.w = __cosf(s1.w * h1.w * inv);
            float* op = out + (size_t)row * (MMB * DD) + m * DD + j0;
            *(float4*)op       = o0;
            *(float4*)(op + 4) = o1;
        }
        // next iteration writes only bufA then barriers -> no extra barrier
    }
}

extern "C" void kernel_launch(void* const* d_in, const int* in_sizes, int n_in,
                              void* d_out, int out_size, void* d_ws, size_t ws_size,
                              hipStream_t stream) {
    const float* x  = (const float*)d_in[0];
    const float* Bw = (const float*)d_in[1];
    const float* Gw = (const float*)d_in[2];
    const float* Sw = (const float*)d_in[3];
    const int*   Pw = (const int*)d_in[4];
    float* out = (float*)d_out;

    fastfood_wmma_kernel<<<NROWS, 128, 0, stream>>>(x, Bw, Gw, Sw, Pw, out);
}